// SMPLSkeleton_61976378082052
// MI455X (gfx1250) — compile-verified
//
#include <hip/hip_runtime.h>
#include <stdint.h>

// ---------------------------------------------------------------------------
// SMPL forward kinematics, B=131072 batches, 24 joints.
//   in : orientations (B,24,3,3) f32 -> 864 B/batch (113 MB)
//   out: positions    (B,24,3)   f32 -> 288 B/batch ( 38 MB)
// Memory bound: ~151 MB @ 23.3 TB/s -> ~6.5 us floor; ~0.09 GFLOP is noise.
// G is a permutation matrix -> all G-einsums are index permutations;
// ANCESTOR matmul == prefix chain pos[j] = pos[par[j]] + delta[j].
//
// CDNA5 paths used:
//  * TENSOR_LOAD_TO_LDS (TDM): one descriptor-driven DMA per block moves the
//    (64 x 216 f32) input tile global->LDS. pad_enable(8dw interval, 1dw pad)
//    gives an effective 243-dword batch stride (odd -> conflict-free LDS
//    reads across a wave). Tracked with TENSORcnt.
//  * GLOBAL_STORE_ASYNC_FROM_LDS_B128: coalesced LDS->global output stream,
//    tracked with ASYNCcnt.
// WMMA deliberately not used: the only matrix work is 3x3 permuted mat-vecs
// and a 24-deep prefix chain (~70 VALU ops/batch) - VALU time is negligible
// against HBM, and WMMA tiling would only add LDS transposes.
// ---------------------------------------------------------------------------

#define TPB        64     // threads per block == batches per block
#define NB         64
#define IN_FL      216    // input floats per batch
#define IN_STRIDE  243    // effective LDS stride after TDM padding (216 + 216/8)
#define OUT_FL     72     // output floats per batch
#define OUT_STRIDE 76     // padded LDS stride for output staging (16B aligned)
#define OUT_CH     18     // 16B chunks of output per batch (72/4)

// LDS dword index of input element e (0..215) after TDM padding:
// one pad dword after every 8 stored dwords.
#define LIDX(e) ((e) + ((e) >> 3))

typedef uint32_t v4u __attribute__((ext_vector_type(4)));
typedef int      v4i __attribute__((ext_vector_type(4)));
typedef int      v8i __attribute__((ext_vector_type(8)));

__device__ __forceinline__ void async_store_b128(const void* g, uint32_t lds_off) {
  // GLOBAL_STORE_ASYNC_FROM_LDS_B128 (GV mode): vaddr = global, vsrc = LDS addr.
  asm volatile("global_store_async_from_lds_b128 %0, %1, off"
               :: "v"(g), "v"(lds_off) : "memory");
}

__device__ __forceinline__ void wait_async_0() {
  asm volatile("s_wait_asynccnt 0x0" ::: "memory");
}

__device__ __forceinline__ void tdm_load_2d(uint32_t lds_addr, const void* gaddr,
                                            int rows /* batches in tile */) {
  const uint64_t ga = (uint64_t)(uintptr_t)gaddr;
  // ---- D# group 0: count=1 | lds_addr | global_addr | type=2 ("image")
  v4u g0;
  g0[0] = 1u;                                   // count=1 (valid), user mode
  g0[1] = lds_addr;                             // LDS byte address
  g0[2] = (uint32_t)ga;                         // global_addr[31:0]
  g0[3] = (uint32_t)(ga >> 32) | (2u << 30);    // global_addr[56:32] | type=2
  // ---- D# group 1:
  //  data_size=2 (4B), pad_enable=1, pad_interval=2 (8 dwords), pad_amount=0 (1 dword)
  //  tensor_dim0=216, tensor_dim1=rows, tile_dim0=216, tile_dim1=rows,
  //  tensor_dim0_stride=216. workgroup_mask=0 (not in a cluster).
  v8i g1;
  g1[0] = (2 << 16) | (1 << 20) | (2 << 22);    // 0x00920000
  g1[1] = (IN_FL << 16);                        // tensor_dim0[15:0] @ bits 63:48
  g1[2] = (rows << 16);                         // tensor_dim1[15:0] @ bits 95:80
  g1[3] = (IN_FL << 16);                        // tile_dim0 @ bits 127:112
  g1[4] = rows;                                 // tile_dim1 @ bits 143:128, tile_dim2=0
  g1[5] = IN_FL;                                // tensor_dim0_stride[31:0]
  g1[6] = 0;
  g1[7] = 0;
  v4i gz = {0, 0, 0, 0};                        // groups 2/3: 2-D tensor
#if defined(__clang_major__) && (__clang_major__ >= 23)
  v8i gz8 = {0, 0, 0, 0, 0, 0, 0, 0};
  __builtin_amdgcn_tensor_load_to_lds(g0, g1, gz, gz, gz8, 0);
#else
  __builtin_amdgcn_tensor_load_to_lds(g0, g1, gz, gz, 0);
#endif
}

__global__ __launch_bounds__(TPB) void smpl_fk_kernel(
    const float* __restrict__ orient,   // (B,24,3,3)
    const float* __restrict__ offsets,  // (24,3)
    float* __restrict__ out,            // (B,24,3)
    int B) {
  // One LDS buffer, reused: TDM-padded input (243 dw/batch), then output
  // staging (76 dw/batch). 64*243*4 = 62208 B.
  __shared__ __align__(16) float tile[NB * IN_STRIDE];
  __shared__ float off_lds[OUT_FL];

  const int tid = threadIdx.x;
  const long tileBase = (long)blockIdx.x * NB;              // first batch of tile
  const int validB = (B - (int)tileBase) < NB ? (B - (int)tileBase) : NB;

  // Low 32 bits of a generic pointer into __shared__ == LDS byte address.
  const uint32_t lds_base = (uint32_t)(uintptr_t)(void*)&tile[0];

  // ---- Phase 1: TDM DMA of the whole input tile. Wave 0 issues it once
  // (TDM ignores EXEC; wave 1 skips the block via execz branch).
  if (tid < 32) {
    tdm_load_2d(lds_base,
                (const char*)orient + (size_t)tileBase * (IN_FL * 4),
                validB);
  }

  // Stage the (uniform, tiny) offsets table while the DMA flies.
  if (tid < OUT_FL) off_lds[tid] = offsets[tid];

  __builtin_amdgcn_s_wait_tensorcnt(0);   // wave1: trivially 0
  __syncthreads();

  // ---- Phase 2: register-resident kinematic chain for this thread's batch.
  // o[j,m] = R[p(j),p(m)], p = (0->2, 1->0, 2->1).
  // delta[k] = R_par[p(k),2]*off0 + R_par[p(k),0]*off1 + R_par[p(k),1]*off2
  //   (rows: k=0 -> row2, k=1 -> row0, k=2 -> row1)
  // pos[j] = pos[par[j]] + delta[j]; pos[0] = 0 (root mask).
  // out[j] = (pos[j].y, pos[j].z, pos[j].x)  (final G permutation).
  constexpr int PAR[24] = {-1, 0, 1, 2, 3, 0, 5, 6, 7, 0, 9, 10, 11, 12,
                           11, 14, 15, 16, 17, 11, 19, 20, 21, 22};
  const float* my = &tile[tid * IN_STRIDE];

  float pos[24][3];
  pos[0][0] = 0.f; pos[0][1] = 0.f; pos[0][2] = 0.f;

  #pragma unroll
  for (int j = 1; j < 24; ++j) {
    const int p = PAR[j];
    const float f0 = off_lds[j * 3 + 0];
    const float f1 = off_lds[j * 3 + 1];
    const float f2 = off_lds[j * 3 + 2];
    // R element (r,c) of parent p lives at padded index LIDX(p*9 + r*3 + c).
    const float r00 = my[LIDX(p * 9 + 0)], r01 = my[LIDX(p * 9 + 1)], r02 = my[LIDX(p * 9 + 2)];
    const float r10 = my[LIDX(p * 9 + 3)], r11 = my[LIDX(p * 9 + 4)], r12 = my[LIDX(p * 9 + 5)];
    const float r20 = my[LIDX(p * 9 + 6)], r21 = my[LIDX(p * 9 + 7)], r22 = my[LIDX(p * 9 + 8)];
    const float d0 = r22 * f0 + r20 * f1 + r21 * f2;
    const float d1 = r02 * f0 + r00 * f1 + r01 * f2;
    const float d2 = r12 * f0 + r10 * f1 + r11 * f2;
    pos[j][0] = pos[p][0] + d0;
    pos[j][1] = pos[p][1] + d1;
    pos[j][2] = pos[p][2] + d2;
  }

  // All LDS input reads done before we overwrite the buffer with outputs.
  __syncthreads();

  // ---- Phase 3: stage permuted outputs into LDS (stride 76 dw = 304 B).
  {
    float* ob = &tile[tid * OUT_STRIDE];
    #pragma unroll
    for (int j = 0; j < 24; ++j) {
      ob[j * 3 + 0] = pos[j][1];
      ob[j * 3 + 1] = pos[j][2];
      ob[j * 3 + 2] = pos[j][0];
    }
  }

  __syncthreads();

  // ---- Phase 4: async DMA output tile LDS -> global (padded -> contiguous).
  {
    char* gout = (char*)out + (size_t)tileBase * (OUT_FL * 4);
    const int nch = validB * OUT_CH;
    #pragma unroll 4
    for (int c = tid; c < NB * OUT_CH; c += TPB) {
      if (c < nch) {
        const int b = c / OUT_CH;
        const int r = c - b * OUT_CH;
        const uint32_t laddr = lds_base + (uint32_t)(b * (OUT_STRIDE * 4) + r * 16);
        async_store_b128(gout + (size_t)(b * (OUT_FL * 4) + r * 16), laddr);
      }
    }
  }

  // Ensure async stores complete while the workgroup (and its LDS) is alive.
  wait_async_0();
}

extern "C" void kernel_launch(void* const* d_in, const int* in_sizes, int n_in,
                              void* d_out, int out_size, void* d_ws, size_t ws_size,
                              hipStream_t stream) {
  const float* orient  = (const float*)d_in[0];   // (B,24,3,3) f32
  const float* offsets = (const float*)d_in[1];   // (24,3) f32
  float* out = (float*)d_out;                     // (B,24,3) f32

  const int B = in_sizes[0] / (24 * 9);
  const int grid = (B + NB - 1) / NB;
  smpl_fk_kernel<<<grid, TPB, 0, stream>>>(orient, offsets, out, B);
}